// Mask2FormerLoss_48670569398856
// MI455X (gfx1250) — compile-verified
//
#include <hip/hip_runtime.h>
#include <hip/hip_bf16.h>
#include <math.h>

typedef float v2f __attribute__((ext_vector_type(2)));
typedef float v8f __attribute__((ext_vector_type(8)));

#define BB    2
#define QQ    100
#define QPAD  112        // 7 * 16
#define NCLS  151
#define GG    20
#define GPAD  32         // 2 * 16
#define HT    256
#define WT    256
#define HWT   (HT*WT)    // 65536
#define HP    128
#define WP    128
#define HWP   (HP*WP)    // 16384
#define CHUNKS 32
#define MTILES 7

#define LOG2E 1.4426950408889634f
#define LN2   0.6931471805599453f

// ---------------------------------------------------------------- fast math
// Raw hardware transcendentals. Safe here: all log arguments are >= 1
// (no denorms / negatives), exp2 saturates gracefully to 0 / +inf.
__device__ inline float fast_exp(float x)   { return __builtin_amdgcn_exp2f(x * LOG2E); }
__device__ inline float fast_log(float x)   { return __builtin_amdgcn_logf(x) * LN2; }
__device__ inline float fast_sigmoid(float x) {
    return __builtin_amdgcn_rcpf(1.f + __builtin_amdgcn_exp2f(-x * LOG2E));
}
// log(sigmoid(z)) = -log(1 + exp(-z))
__device__ inline float fast_logsig(float z) { return -fast_log(1.f + fast_exp(-z)); }

// ---------------------------------------------------------------- utilities

__global__ void zero_kernel(float* __restrict__ p, int n) {
    int i = blockIdx.x * blockDim.x + threadIdx.x;
    if (i < n) p[i] = 0.f;
}

__global__ void init_out_kernel(float* __restrict__ out) {
    if (threadIdx.x == 0 && blockIdx.x == 0) out[0] = 0.f;
}

__device__ inline float block_reduce_sum(float v, float* sm) {
    int t = threadIdx.x;
    sm[t] = v; __syncthreads();
    for (int s = blockDim.x >> 1; s > 0; s >>= 1) {
        if (t < s) sm[t] += sm[t + s];
        __syncthreads();
    }
    float r = sm[0]; __syncthreads();
    return r;
}

// ------------------------------------------------- resize + sigmoid (128->256)
// jax.image.resize bilinear, half-pixel centers; upsample => plain 2-tap bilerp.
__global__ __launch_bounds__(256) void resize_sigmoid_kernel(
        const float* __restrict__ masks, float* __restrict__ pm) {
    int idx = blockIdx.x * 256 + threadIdx.x;          // B*Q*65536 = 13107200
    int x = idx & 255;
    int y = (idx >> 8) & 255;
    int bq = idx >> 16;
    const float* src = masks + (size_t)bq * HWP;
    float fy = 0.5f * (float)y - 0.25f;
    float fx = 0.5f * (float)x - 0.25f;
    int y0 = (int)floorf(fy); float wy = fy - (float)y0;
    int x0 = (int)floorf(fx); float wx = fx - (float)x0;
    int y0c = min(max(y0, 0), HP - 1), y1c = min(max(y0 + 1, 0), HP - 1);
    int x0c = min(max(x0, 0), WP - 1), x1c = min(max(x0 + 1, 0), WP - 1);
    float v = (1.f - wy) * ((1.f - wx) * src[y0c * WP + x0c] + wx * src[y0c * WP + x1c])
            +         wy * ((1.f - wx) * src[y1c * WP + x0c] + wx * src[y1c * WP + x1c]);
    pm[idx] = fast_sigmoid(v);
}

// -------------------------------------------- gt downsample 256->128 (antialias)
// jax.image.resize bilinear w/ antialias on 2x downscale = separable [1,3,3,1]/8.
__global__ __launch_bounds__(256) void gt_resize_kernel(
        const float* __restrict__ T, float* __restrict__ gt) {
    int idx = blockIdx.x * 256 + threadIdx.x;          // B*G*16384
    if (idx >= BB * GG * HWP) return;
    int x = idx & 127;
    int y = (idx >> 7) & 127;
    int bg = idx >> 14;
    const float* src = T + (size_t)bg * HWT;
    const float w[4] = {0.125f, 0.375f, 0.375f, 0.125f};
    float acc = 0.f;
    #pragma unroll
    for (int dy = 0; dy < 4; ++dy) {
        int iy = min(max(2 * y - 1 + dy, 0), HT - 1);
        float ry = 0.f;
        #pragma unroll
        for (int dx = 0; dx < 4; ++dx) {
            int ix = min(max(2 * x - 1 + dx, 0), WT - 1);
            ry += w[dx] * src[iy * WT + ix];
        }
        acc += w[dy] * ry;
    }
    gt[idx] = acc;
}

// ------------------------------------------------------------ T row sums (256^2)
__global__ __launch_bounds__(256) void tsum_kernel(
        const float* __restrict__ T, float* __restrict__ tsum) {
    __shared__ float sm[256];
    int bg = blockIdx.x;
    const float* row = T + (size_t)bg * HWT;
    float s = 0.f;
    for (int i = threadIdx.x; i < HWT; i += 256) s += row[i];
    s = block_reduce_sum(s, sm);
    if (threadIdx.x == 0) tsum[bg] = s;
}

// ------------------------------------------------------- class cost (softmax)
__global__ __launch_bounds__(256) void cls_cost_kernel(
        const float* __restrict__ logits, const int* __restrict__ labels,
        float* __restrict__ clsc) {
    __shared__ float sm[256];
    int bq = blockIdx.x;                                // b*Q + q
    int b = bq / QQ;
    const float* lg = logits + (size_t)bq * NCLS;
    int t = threadIdx.x;
    float xv = (t < NCLS) ? lg[t] : -3.0e38f;
    sm[t] = xv; __syncthreads();
    for (int s = 128; s > 0; s >>= 1) { if (t < s) sm[t] = fmaxf(sm[t], sm[t + s]); __syncthreads(); }
    float mx = sm[0]; __syncthreads();
    float e = (t < NCLS) ? fast_exp(lg[t] - mx) : 0.f;
    sm[t] = e; __syncthreads();
    for (int s = 128; s > 0; s >>= 1) { if (t < s) sm[t] += sm[t + s]; __syncthreads(); }
    float den = sm[0];
    if (t < GG) {
        int lbl = labels[b * GG + t];
        clsc[(size_t)bq * GG + t] = -fast_exp(lg[lbl] - mx) / den;
    }
}

// --------------------------------------------------- WMMA cost-matrix GEMMs
// A1 = sum_h logsig(pm)  * T      (per b: [QPAD x GPAD])
// A2 = sum_h logsig(-pm) * T
// A3 = sum_h pm * T
// R2[q] = sum_h logsig(-pm),  R3[q] = sum_h pm
__global__ __launch_bounds__(128) void cost_wmma_kernel(
        const float* __restrict__ pm, const float* __restrict__ T,
        float* __restrict__ A1, float* __restrict__ A2, float* __restrict__ A3,
        float* __restrict__ R2, float* __restrict__ R3) {
    const int bid   = blockIdx.x;
    const int b     = bid / (MTILES * CHUNKS);
    const int rem   = bid % (MTILES * CHUNKS);
    const int mt    = rem / CHUNKS;
    const int chunk = rem % CHUNKS;
    const int wave  = threadIdx.x >> 5;
    const int lane  = threadIdx.x & 31;
    const int row   = lane & 15;    // A: M within tile | B: N within tile
    const int khalf = lane >> 4;    // k-pair select {0,1} vs {2,3}

    const int span  = HWT / CHUNKS / 4;                 // elems per wave
    const int hbase = chunk * (HWT / CHUNKS) + wave * span;

    const int q = mt * 16 + row;
    const float* __restrict__ Arow  = pm + ((size_t)b * QQ + (q < QQ ? q : 0)) * HWT;
    const float* __restrict__ Brow0 = T  + ((size_t)b * GG + row) * HWT;          // row < 16 < G
    const int g1 = 16 + row;
    const float* __restrict__ Brow1 = T  + ((size_t)b * GG + (g1 < GG ? g1 : 0)) * HWT;

    v8f a1_0 = {}, a1_1 = {}, a2_0 = {}, a2_1 = {}, a3_0 = {}, a3_1 = {};
    float r2 = 0.f, r3 = 0.f;

    for (int s = 0; s < span; s += 4) {
        const int h = hbase + s + 2 * khalf;
        float p0 = Arow[h], p1 = Arow[h + 1];
        // log-sigmoid of pm and -pm; pm in (0,1) => log args in (1, 3.72): safe
        float x1a = fast_logsig(p0);
        float x1b = fast_logsig(p1);
        float x2a = fast_logsig(-p0);
        float x2b = fast_logsig(-p1);
        r2 += x2a + x2b;
        r3 += p0 + p1;
        v2f fa1 = {x1a, x1b}, fa2 = {x2a, x2b}, fa3 = {p0, p1};
        v2f fb0 = {Brow0[h], Brow0[h + 1]};
        v2f fb1 = {Brow1[h], Brow1[h + 1]};
        a1_0 = __builtin_amdgcn_wmma_f32_16x16x4_f32(false, fa1, false, fb0, (short)0, a1_0, false, false);
        a1_1 = __builtin_amdgcn_wmma_f32_16x16x4_f32(false, fa1, false, fb1, (short)0, a1_1, false, false);
        a2_0 = __builtin_amdgcn_wmma_f32_16x16x4_f32(false, fa2, false, fb0, (short)0, a2_0, false, false);
        a2_1 = __builtin_amdgcn_wmma_f32_16x16x4_f32(false, fa2, false, fb1, (short)0, a2_1, false, false);
        a3_0 = __builtin_amdgcn_wmma_f32_16x16x4_f32(false, fa3, false, fb0, (short)0, a3_0, false, false);
        a3_1 = __builtin_amdgcn_wmma_f32_16x16x4_f32(false, fa3, false, fb1, (short)0, a3_1, false, false);
    }

    // C/D layout: vgpr r, lanes0-15 -> M=r, lanes16-31 -> M=r+8 ; N = lane&15
    const int mbase = mt * 16 + 8 * khalf;
    #pragma unroll
    for (int r = 0; r < 8; ++r) {
        const size_t o0 = ((size_t)b * QPAD + mbase + r) * GPAD + row;
        const size_t o1 = o0 + 16;
        atomicAdd(&A1[o0], a1_0[r]); atomicAdd(&A1[o1], a1_1[r]);
        atomicAdd(&A2[o0], a2_0[r]); atomicAdd(&A2[o1], a2_1[r]);
        atomicAdd(&A3[o0], a3_0[r]); atomicAdd(&A3[o1], a3_1[r]);
    }
    atomicAdd(&R2[(size_t)b * QPAD + mt * 16 + row], r2);
    atomicAdd(&R3[(size_t)b * QPAD + mt * 16 + row], r3);
}

// ------------------------------------------------------------- assemble C[b,q,g]
__global__ __launch_bounds__(256) void combine_cost_kernel(
        const float* __restrict__ clsc,
        const float* __restrict__ A1, const float* __restrict__ A2,
        const float* __restrict__ A3, const float* __restrict__ R2,
        const float* __restrict__ R3, const float* __restrict__ tsum,
        float* __restrict__ C) {
    int idx = blockIdx.x * 256 + threadIdx.x;           // B*Q*G = 4000
    if (idx >= BB * QQ * GG) return;
    int g = idx % GG;
    int q = (idx / GG) % QQ;
    int b = idx / (QQ * GG);
    size_t ap = ((size_t)b * QPAD + q) * GPAD + g;
    float a1 = A1[ap], a2 = A2[ap], a3 = A3[ap];
    float r2 = R2[b * QPAD + q], r3 = R3[b * QPAD + q];
    float bce  = (-a1 - (r2 - a2)) / (float)HWT;
    float dice = 1.f - (2.f * a3 + 1.f) / (r3 + tsum[b * GG + g] + 1.f);
    C[idx] = 1.f * clsc[idx] + 5.f * bce + 2.f * dice;
}

// --------------------------------------------------- Hungarian (rows=G, cols=Q)
__global__ void hungarian_kernel(const float* __restrict__ C, int* __restrict__ qidx) {
    if (threadIdx.x != 0) return;
    const int b = blockIdx.x;
    const int n = GG, m = QQ;
    double u[GG + 1], v[QQ + 1], minv[QQ + 1];
    int p[QQ + 1], way[QQ + 1];
    bool used[QQ + 1];
    for (int i = 0; i <= n; ++i) u[i] = 0.0;
    for (int j = 0; j <= m; ++j) { v[j] = 0.0; p[j] = 0; }
    for (int i = 1; i <= n; ++i) {
        p[0] = i;
        int j0 = 0;
        for (int j = 0; j <= m; ++j) { minv[j] = 1e18; used[j] = false; way[j] = 0; }
        do {
            used[j0] = true;
            int i0 = p[j0], j1 = -1;
            double delta = 1e18;
            for (int j = 1; j <= m; ++j) if (!used[j]) {
                // cost[i0-1][j-1] = C[b][q=j-1][g=i0-1]
                double cur = (double)C[((size_t)b * QQ + (j - 1)) * GG + (i0 - 1)] - u[i0] - v[j];
                if (cur < minv[j]) { minv[j] = cur; way[j] = j0; }
                if (minv[j] < delta) { delta = minv[j]; j1 = j; }
            }
            for (int j = 0; j <= m; ++j) {
                if (used[j]) { u[p[j]] += delta; v[j] -= delta; }
                else          minv[j] -= delta;
            }
            j0 = j1;
        } while (p[j0] != 0);
        while (j0) { int j1 = way[j0]; p[j0] = p[j1]; j0 = j1; }
    }
    for (int j = 1; j <= m; ++j)
        if (p[j] > 0) qidx[b * GG + p[j] - 1] = j - 1;
}

// ------------------------------------------------------------ weighted CE loss
__global__ __launch_bounds__(256) void ce_kernel(
        const float* __restrict__ logits, const int* __restrict__ labels,
        const int* __restrict__ qidx, const float* __restrict__ ew,
        float* __restrict__ acc) {
    __shared__ float sm[256];
    int bq = blockIdx.x;
    int b = bq / QQ, q = bq % QQ;
    const float* lg = logits + (size_t)bq * NCLS;
    int t = threadIdx.x;
    float xv = (t < NCLS) ? lg[t] : -3.0e38f;
    sm[t] = xv; __syncthreads();
    for (int s = 128; s > 0; s >>= 1) { if (t < s) sm[t] = fmaxf(sm[t], sm[t + s]); __syncthreads(); }
    float mx = sm[0]; __syncthreads();
    float e = (t < NCLS) ? fast_exp(lg[t] - mx) : 0.f;
    sm[t] = e; __syncthreads();
    for (int s = 128; s > 0; s >>= 1) { if (t < s) sm[t] += sm[t + s]; __syncthreads(); }
    float den = sm[0];
    if (t == 0) {
        int c = NCLS - 1;                               // "no object"
        for (int g = 0; g < GG; ++g)
            if (qidx[b * GG + g] == q) c = labels[b * GG + g];
        float nll = (mx + fast_log(den)) - lg[c];
        float w = ew[c];
        atomicAdd(&acc[0], w * nll);
        atomicAdd(&acc[1], w);
    }
}

// ------------------------------------------------- matched BCE + dice (128^2)
__global__ __launch_bounds__(256) void mask_loss_kernel(
        const float* __restrict__ masks, const float* __restrict__ gt,
        const int* __restrict__ qidx, float* __restrict__ acc) {
    __shared__ float sm[256];
    int bg = blockIdx.x;
    int b = bg / GG;
    int qs = qidx[bg];
    const float* x = masks + ((size_t)b * QQ + qs) * HWP;
    const float* t = gt + (size_t)bg * HWP;
    float sb = 0.f, spt = 0.f, sp = 0.f, st = 0.f;
    for (int i = threadIdx.x; i < HWP; i += 256) {
        float xv = x[i], tv = t[i];
        // softplus(-|x|): log arg in (1, 2]: safe for raw hw log
        sb += fmaxf(xv, 0.f) - xv * tv + fast_log(1.f + fast_exp(-fabsf(xv)));
        float pv = fast_sigmoid(xv);
        spt += pv * tv; sp += pv; st += tv;
    }
    sb  = block_reduce_sum(sb,  sm);
    spt = block_reduce_sum(spt, sm);
    sp  = block_reduce_sum(sp,  sm);
    st  = block_reduce_sum(st,  sm);
    if (threadIdx.x == 0) {
        atomicAdd(&acc[2], sb);
        float dice = 1.f - (2.f * spt + 1.f) / (sp + st + 1.f);
        atomicAdd(&acc[3], dice);
    }
}

// ------------------------------------------------------------------ finalize
__global__ void finalize_kernel(const float* __restrict__ acc, float* __restrict__ out) {
    if (threadIdx.x == 0 && blockIdx.x == 0) {
        float ce  = acc[0] / acc[1];
        float bce = acc[2] / ((float)GG * (float)HWP) / (float)(BB * GG);
        float dl  = acc[3] / (float)GG / (float)(BB * GG);
        out[0] += 2.f * ce + 5.f * bce + 5.f * dl;
    }
}

// ================================================================= launcher
extern "C" void kernel_launch(void* const* d_in, const int* in_sizes, int n_in,
                              void* d_out, int out_size, void* d_ws, size_t ws_size,
                              hipStream_t stream) {
    const float* pred_logits = (const float*)d_in[0];   // [2,100,151]
    const float* pred_masks  = (const float*)d_in[1];   // [2,100,128,128]
    const float* aux_logits  = (const float*)d_in[2];   // [2,2,100,151]
    const float* aux_masks   = (const float*)d_in[3];   // [2,2,100,128,128]
    const int*   tgt_labels  = (const int*)  d_in[4];   // [2,20]
    const float* tgt_masks   = (const float*)d_in[5];   // [2,20,256,256]
    const float* empty_w     = (const float*)d_in[6];   // [151]
    float* out = (float*)d_out;
    float* ws  = (float*)d_ws;
    (void)in_sizes; (void)n_in; (void)out_size; (void)ws_size;

    // workspace layout (floats)
    const size_t o_pm   = 0;                                  // B*Q*HWT
    const size_t o_gt   = o_pm   + (size_t)BB * QQ * HWT;     // B*G*HWP
    const size_t o_clsc = o_gt   + (size_t)BB * GG * HWP;     // B*Q*G
    const size_t o_C    = o_clsc + (size_t)BB * QQ * GG;      // B*Q*G
    const size_t o_tsum = o_C    + (size_t)BB * QQ * GG;      // B*G
    const size_t o_qidx = o_tsum + (size_t)BB * GG;           // B*G (ints)
    const size_t o_A1   = o_qidx + (size_t)BB * GG;           // zero block start
    const size_t o_A2   = o_A1   + (size_t)BB * QPAD * GPAD;
    const size_t o_A3   = o_A2   + (size_t)BB * QPAD * GPAD;
    const size_t o_R2   = o_A3   + (size_t)BB * QPAD * GPAD;
    const size_t o_R3   = o_R2   + (size_t)BB * QPAD;
    const size_t o_acc  = o_R3   + (size_t)BB * QPAD;
    const int    zn     = (int)(o_acc + 4 - o_A1);            // zero-block floats

    int* qidx = (int*)(ws + o_qidx);

    init_out_kernel<<<1, 32, 0, stream>>>(out);
    tsum_kernel<<<BB * GG, 256, 0, stream>>>(tgt_masks, ws + o_tsum);
    gt_resize_kernel<<<(BB * GG * HWP + 255) / 256, 256, 0, stream>>>(tgt_masks, ws + o_gt);

    for (int l = 0; l < 3; ++l) {
        const float* lg = (l == 0) ? pred_logits
                                   : aux_logits + (size_t)(l - 1) * BB * QQ * NCLS;
        const float* mk = (l == 0) ? pred_masks
                                   : aux_masks + (size_t)(l - 1) * BB * QQ * HWP;

        zero_kernel<<<(zn + 255) / 256, 256, 0, stream>>>(ws + o_A1, zn);
        resize_sigmoid_kernel<<<(BB * QQ * HWT) / 256, 256, 0, stream>>>(mk, ws + o_pm);
        cls_cost_kernel<<<BB * QQ, 256, 0, stream>>>(lg, tgt_labels, ws + o_clsc);
        cost_wmma_kernel<<<BB * MTILES * CHUNKS, 128, 0, stream>>>(
            ws + o_pm, tgt_masks,
            ws + o_A1, ws + o_A2, ws + o_A3, ws + o_R2, ws + o_R3);
        combine_cost_kernel<<<(BB * QQ * GG + 255) / 256, 256, 0, stream>>>(
            ws + o_clsc, ws + o_A1, ws + o_A2, ws + o_A3,
            ws + o_R2, ws + o_R3, ws + o_tsum, ws + o_C);
        hungarian_kernel<<<BB, 32, 0, stream>>>(ws + o_C, qidx);
        ce_kernel<<<BB * QQ, 256, 0, stream>>>(lg, tgt_labels, qidx, empty_w, ws + o_acc);
        mask_loss_kernel<<<BB * GG, 256, 0, stream>>>(mk, ws + o_gt, qidx, ws + o_acc);
        finalize_kernel<<<1, 32, 0, stream>>>(ws + o_acc, out);
    }
}